// GLR_49701361549669
// MI455X (gfx1250) — compile-verified
//
#include <hip/hip_runtime.h>

// ---------------------------------------------------------------------------
// GLR forward for MI455X (gfx1250, wave32).
// CNN stages: direct f32 conv kernels (tiny fraction of FLOPs).
// Dense solve (dominant, ~11.6 GFLOP): blocked LU without pivoting on the
// augmented [A|y] matrix using V_WMMA_F32_16X16X4_F32 for panel + trailing
// updates. Whole 54MB matrix set is L2-resident on MI455X (192MB L2).
// ---------------------------------------------------------------------------

typedef float v2f __attribute__((ext_vector_type(2)));
typedef float v8f __attribute__((ext_vector_type(8)));

#define NPIX 1296        // 36*36
#define LDA  1312        // 82 tiles of 16 (1296 cols of A + 16 augmented)
#define BATCH 8
#define U_CLIP 15.5625f  // (250-1)/(2*8)

// ------------------------------ conv ---------------------------------------
// Generic 3x3 conv, pad=1. Handles: stride 1/2, optional relu, deconv weight
// indexing (w[ci][co][2-r][2-s]), up2-folded input (logical HxW, stored H/2 x
// W/2), dual-source input (channel concat), optional residual add.
__global__ __launch_bounds__(256) void conv3x3_kernel(
    const float* __restrict__ in1, int C1,
    const float* __restrict__ in2, int C2,
    const float* __restrict__ w, const float* __restrict__ bias,
    const float* __restrict__ residual,
    float* __restrict__ out,
    int Cout, int Hin, int Win, int Hout, int Wout,
    int stride, int relu_f, int deconv_f, int up2_f)
{
    int pix = blockIdx.x * blockDim.x + threadIdx.x;
    int total = Hout * Wout;
    if (pix >= total) return;
    int co = blockIdx.y;
    int b  = blockIdx.z;
    int oy = pix / Wout, ox = pix % Wout;
    int Cin = C1 + C2;
    int Hs = up2_f ? (Hin >> 1) : Hin;
    int Ws = up2_f ? (Win >> 1) : Win;

    float acc = bias[co];
    for (int ci = 0; ci < Cin; ++ci) {
        const float* src = (ci < C1)
            ? (in1 + ((size_t)b * C1 + ci) * (size_t)(Hs * Ws))
            : (in2 + ((size_t)b * C2 + (ci - C1)) * (size_t)(Hs * Ws));
        for (int r = 0; r < 3; ++r) {
            int iy = oy * stride + r - 1;
            if (iy < 0 || iy >= Hin) continue;
            int sy = up2_f ? (iy >> 1) : iy;
            for (int s = 0; s < 3; ++s) {
                int ix = ox * stride + s - 1;
                if (ix < 0 || ix >= Win) continue;
                int sx = up2_f ? (ix >> 1) : ix;
                float wv;
                if (deconv_f)
                    wv = w[(((size_t)ci * Cout + co) * 3 + (2 - r)) * 3 + (2 - s)];
                else
                    wv = w[(((size_t)co * Cin + ci) * 3 + r) * 3 + s];
                acc = fmaf(src[sy * Ws + sx], wv, acc);
            }
        }
    }
    if (residual) acc += residual[(((size_t)b * Cout + co) * total) + pix];
    if (relu_f) acc = fmaxf(acc, 0.0f);
    out[(((size_t)b * Cout + co) * total) + pix] = acc;
}

// ------------------------------ maxpool 2x2 ---------------------------------
__global__ __launch_bounds__(256) void maxpool2_kernel(
    const float* __restrict__ in, float* __restrict__ out,
    int C, int H, int W, int Ho, int Wo)
{
    int idx = blockIdx.x * blockDim.x + threadIdx.x;
    int total = BATCH * C * Ho * Wo;
    if (idx >= total) return;
    int wo = idx % Wo; int t = idx / Wo;
    int ho = t % Ho;  t /= Ho;
    int c  = t % C;   int b = t / C;
    const float* src = in + ((size_t)b * C + c) * (size_t)(H * W);
    float m = -3.0e38f;                 // ceil_mode pads with -inf: just skip OOB
    for (int dy = 0; dy < 2; ++dy) {
        int y = 2 * ho + dy; if (y >= H) continue;
        for (int dx = 0; dx < 2; ++dx) {
            int x = 2 * wo + dx; if (x >= W) continue;
            m = fmaxf(m, src[y * W + x]);
        }
    }
    out[idx] = m;
}

// ------------------------------ fully connected -----------------------------
__global__ __launch_bounds__(256) void fc_kernel(
    const float* __restrict__ in, const float* __restrict__ w,
    const float* __restrict__ bias, float* __restrict__ out,
    int In, int Out, int clip_f)
{
    int idx = blockIdx.x * blockDim.x + threadIdx.x;
    if (idx >= BATCH * Out) return;
    int b = idx / Out, o = idx % Out;
    float acc = bias[o];
    for (int k = 0; k < In; ++k)
        acc = fmaf(in[(size_t)b * In + k], w[(size_t)o * In + k], acc);
    if (clip_f) acc = fminf(acc, U_CLIP);
    out[idx] = acc;
}

// ------------------------------ build augmented A ---------------------------
// A_ij = delta_ij + u*(delta_ij*deg_i - W_ij), W = exp(-dist), augmented col
// 1296 = y, cols 1297..1311 = 0. One 256-thread block per (row, batch).
__global__ __launch_bounds__(256) void build_A_kernel(
    const float* __restrict__ E, const float* __restrict__ Yv,
    const float* __restrict__ u, float* __restrict__ A)
{
    int i = blockIdx.x;
    int b = blockIdx.y;
    const float* Eb = E + (size_t)b * 3 * NPIX;
    float e0 = Eb[i], e1 = Eb[NPIX + i], e2 = Eb[2 * NPIX + i];
    float sqi = e0 * e0 + e1 * e1 + e2 * e2;
    float ub = u[b];
    float* Arow = A + ((size_t)b * NPIX + i) * LDA;

    __shared__ float red[256];
    float degp = 0.0f;
    for (int j = threadIdx.x; j < NPIX; j += 256) {
        float f0 = Eb[j], f1 = Eb[NPIX + j], f2 = Eb[2 * NPIX + j];
        float sqj = f0 * f0 + f1 * f1 + f2 * f2;
        float dist = sqi + sqj - 2.0f * (e0 * f0 + e1 * f1 + e2 * f2);
        float wv = __expf(-dist);
        degp += wv;
        Arow[j] = -ub * wv;
    }
    red[threadIdx.x] = degp;
    __syncthreads();
    for (int s = 128; s > 0; s >>= 1) {
        if (threadIdx.x < s) red[threadIdx.x] += red[threadIdx.x + s];
        __syncthreads();
    }
    if (threadIdx.x == 0) {
        Arow[i] += 1.0f + ub * red[0];        // A_ii = 1 + u*(deg_i - W_ii)
        Arow[NPIX] = Yv[(size_t)b * NPIX + i];
    }
    if (threadIdx.x >= 1 && threadIdx.x <= 15) Arow[NPIX + threadIdx.x] = 0.0f;
}

// ------------------------------ 16x16 diagonal block inverse ----------------
// A is strictly diagonally dominant => Gauss-Jordan without pivoting is safe.
__global__ __launch_bounds__(256) void invert16_kernel(
    const float* __restrict__ A, float* __restrict__ Dinv, int k)
{
    int b = blockIdx.x;
    __shared__ float M[16][16];
    __shared__ float Inv[16][17];
    __shared__ float sPiv;
    int t = threadIdx.x, r = t >> 4, c = t & 15;
    const float* base = A + ((size_t)b * NPIX + 16 * k) * LDA + 16 * k;
    M[r][c]   = base[(size_t)r * LDA + c];
    Inv[r][c] = (r == c) ? 1.0f : 0.0f;
    __syncthreads();
    for (int p = 0; p < 16; ++p) {
        if (t == 0) sPiv = 1.0f / M[p][p];
        __syncthreads();
        if (r == p) { M[p][c] *= sPiv; Inv[p][c] *= sPiv; }
        __syncthreads();
        float f = M[r][p];
        float mpc = M[p][c], ipc = Inv[p][c];
        __syncthreads();
        if (r != p) { M[r][c] -= f * mpc; Inv[r][c] -= f * ipc; }
        __syncthreads();
    }
    Dinv[((size_t)b * 81 + k) * 256 + r * 16 + c] = Inv[r][c];
}

// ------------------------------ panel: P = -(A_sub * Dinv) ------------------
// One wave per 16x16 tile. WMMA f32 16x16x4, K=16 via 4 chained ops.
__global__ __launch_bounds__(32) void panel_kernel(
    float* __restrict__ A, const float* __restrict__ Dinv, int k)
{
    int rt   = k + 1 + blockIdx.x;    // row tile
    int b    = blockIdx.y;
    int lane = threadIdx.x;
    int hv   = lane >> 4;             // 0: lanes 0-15, 1: lanes 16-31
    int l16  = lane & 15;

    float* Ab = A + (size_t)b * NPIX * LDA;
    const float* Dv = Dinv + ((size_t)b * 81 + k) * 256;
    float* P = Ab + (size_t)(16 * rt) * LDA + 16 * k;

    // A operand (16x4 slices): lanes 0-15 -> K={4j,4j+1}; lanes 16-31 -> K={4j+2,4j+3}
    v2f a[4];
#pragma unroll
    for (int j = 0; j < 4; ++j) {
        const float* pr = P + (size_t)l16 * LDA + 4 * j + 2 * hv;
        a[j][0] = pr[0]; a[j][1] = pr[1];
    }
    v8f acc = {};
#pragma unroll
    for (int j = 0; j < 4; ++j) {
        int kr = 4 * j + 2 * hv;      // B operand (4x16): rows kr, kr+1 at col l16
        v2f bv; bv[0] = Dv[kr * 16 + l16]; bv[1] = Dv[(kr + 1) * 16 + l16];
        acc = __builtin_amdgcn_wmma_f32_16x16x4_f32(
            false, a[j], false, bv, (short)0, acc, false, false);
    }
    // C/D layout: VGPR r -> row r (lanes 0-15) / row r+8 (lanes 16-31)
#pragma unroll
    for (int r = 0; r < 8; ++r)
        P[(size_t)(r + 8 * hv) * LDA + l16] = -acc[r];
}

// ------------------------------ trailing update: C += P * U -----------------
// One wave computes a 16x64 strip (A-operand reused across 4 column tiles).
__global__ __launch_bounds__(32) void trailing_kernel(float* __restrict__ A, int k)
{
    int ctg  = blockIdx.x;            // column tile group of 4
    int rt   = k + 1 + blockIdx.y;    // row tile
    int b    = blockIdx.z;
    int lane = threadIdx.x;
    int hv   = lane >> 4;
    int l16  = lane & 15;

    float* Ab = A + (size_t)b * NPIX * LDA;
    const float* P    = Ab + (size_t)(16 * rt) * LDA + 16 * k;  // holds -L
    const float* Urow = Ab + (size_t)(16 * k) * LDA;            // pivot block row

    v2f a[4];
#pragma unroll
    for (int j = 0; j < 4; ++j) {
        const float* pr = P + (size_t)l16 * LDA + 4 * j + 2 * hv;
        a[j][0] = pr[0]; a[j][1] = pr[1];
    }

    int ct0 = k + 1 + ctg * 4;
    for (int cc = 0; cc < 4; ++cc) {
        int ct = ct0 + cc;
        if (ct > 81) break;           // uniform across the wave (EXEC stays full)
        const float* U = Urow + 16 * ct;
        float* C = Ab + (size_t)(16 * rt) * LDA + 16 * ct;
        v8f acc;
#pragma unroll
        for (int r = 0; r < 8; ++r)
            acc[r] = C[(size_t)(r + 8 * hv) * LDA + l16];
#pragma unroll
        for (int j = 0; j < 4; ++j) {
            int kr = 4 * j + 2 * hv;
            v2f bv; bv[0] = U[(size_t)kr * LDA + l16];
                    bv[1] = U[(size_t)(kr + 1) * LDA + l16];
            acc = __builtin_amdgcn_wmma_f32_16x16x4_f32(
                false, a[j], false, bv, (short)0, acc, false, false);
        }
#pragma unroll
        for (int r = 0; r < 8; ++r)
            C[(size_t)(r + 8 * hv) * LDA + l16] = acc[r];
    }
}

// ------------------------------ back substitution ---------------------------
// One 256-thread block per batch; x kept in LDS; 16 rows x 16 partial dots.
__global__ __launch_bounds__(256) void backsub_kernel(
    const float* __restrict__ A, const float* __restrict__ Dinv,
    float* __restrict__ out)
{
    int b = blockIdx.x;
    const float* Ab = A + (size_t)b * NPIX * LDA;
    __shared__ float x[NPIX];
    __shared__ float tvec[16];
    __shared__ float red[16][17];
    int tid = threadIdx.x;
    int r = tid >> 4, g = tid & 15;

    for (int k = 80; k >= 0; --k) {
        const float* Arow = Ab + (size_t)(16 * k + r) * LDA;
        float s = 0.0f;
        for (int j = 16 * (k + 1) + g; j < NPIX; j += 16)
            s += Arow[j] * x[j];
        red[r][g] = s;
        __syncthreads();
        if (g == 0) {
            float tot = 0.0f;
            for (int q = 0; q < 16; ++q) tot += red[r][q];
            tvec[r] = Arow[NPIX] - tot;
        }
        __syncthreads();
        if (g == 0) {
            const float* Dv = Dinv + ((size_t)b * 81 + k) * 256 + r * 16;
            float xv = 0.0f;
            for (int c = 0; c < 16; ++c) xv += Dv[c] * tvec[c];
            x[16 * k + r] = xv;
        }
        __syncthreads();
    }
    for (int i = tid; i < NPIX; i += 256)
        out[(size_t)b * NPIX + i] = x[i];
}

// ---------------------------------------------------------------------------
extern "C" void kernel_launch(void* const* d_in, const int* in_sizes, int n_in,
                              void* d_out, int out_size, void* d_ws, size_t ws_size,
                              hipStream_t stream) {
    (void)in_sizes; (void)n_in; (void)out_size; (void)ws_size;
    const float* IN[47];
    for (int i = 0; i < 47; ++i) IN[i] = (const float*)d_in[i];
    // 0:x  1:f1a_w 2:f1a_b 3:f1b_w 4:f1b_b 5:f2a_w 6:f2a_b 7:f2_w 8:f2_b
    // 9:f3a_w 10:f3a_b 11:f3_w 12:f3_b 13:d1_w 14:d1_b 15:f4a_w 16:f4a_b
    // 17:f4b_w 18:f4b_b 19:d2_w 20:d2_b 21:f5a_w 22:f5a_b 23:f5b_w 24:f5b_b
    // 25:f5c_w 26:f5c_b 27:y1_w 28:y1_b 29:y2_w 30:y2_b 31:y3_w 32:y3_b
    // 33:y4_w 34:y4_b 35:u1_w 36:u1_b 37:u2_w 38:u2_b 39:u3_w 40:u3_b
    // 41:u4_w 42:u4_b 43:fc1_w 44:fc1_b 45:fc2_w 46:fc2_b
    const float* x = IN[0];

    // ---- workspace bump allocator (deterministic layout; needs ~75 MB) ----
    float* wp = (float*)d_ws;
    auto alloc = [&](size_t n) { float* r = wp; wp += n; return r; };
    float* t1   = alloc((size_t)8 * 32 * 1296);
    float* o1   = alloc((size_t)8 * 32 * 1296);
    float* o2   = alloc((size_t)8 * 64 * 1296);
    float* o2m  = alloc((size_t)8 * 64 * 324);
    float* o2p  = alloc((size_t)8 * 64 * 324);
    float* o3   = alloc((size_t)8 * 64 * 324);
    float* o3m  = alloc((size_t)8 * 64 * 81);
    float* o3p  = alloc((size_t)8 * 128 * 81);
    float* o3u  = alloc((size_t)8 * 64 * 324);
    float* o4a  = alloc((size_t)8 * 64 * 324);
    float* o4   = alloc((size_t)8 * 64 * 324);
    float* o4u  = alloc((size_t)8 * 32 * 1296);
    float* f5t1 = alloc((size_t)8 * 32 * 1296);
    float* f5t2 = alloc((size_t)8 * 32 * 1296);
    float* Ebuf = alloc((size_t)8 * 3 * 1296);
    float* yt1  = alloc((size_t)8 * 32 * 1296);
    float* yt2  = alloc((size_t)8 * 32 * 1296);
    float* Yv   = alloc((size_t)8 * 1296);
    float* cu1  = alloc((size_t)8 * 32 * 324);
    float* cu2  = alloc((size_t)8 * 32 * 324);
    float* cu2m = alloc((size_t)8 * 32 * 81);
    float* cu3  = alloc((size_t)8 * 32 * 81);
    float* cu3m = alloc((size_t)8 * 32 * 25);
    float* cu4  = alloc((size_t)8 * 32 * 25);
    float* cu4m = alloc((size_t)8 * 32 * 9);
    float* fco  = alloc((size_t)8 * 32);
    float* uu   = alloc((size_t)8);
    float* Amat = alloc((size_t)8 * NPIX * LDA);
    float* Dinv = alloc((size_t)8 * 81 * 256);

    auto conv = [&](const float* in1, int C1, const float* in2, int C2,
                    int wi, const float* res, float* out, int Cout,
                    int Hin, int Win, int Hout, int Wout,
                    int stride, int relu, int deconv, int up2) {
        dim3 g((Hout * Wout + 255) / 256, Cout, BATCH);
        conv3x3_kernel<<<g, 256, 0, stream>>>(in1, C1, in2, C2, IN[wi], IN[wi + 1],
                                              res, out, Cout, Hin, Win, Hout, Wout,
                                              stride, relu, deconv, up2);
    };
    auto pool = [&](const float* in, float* out, int C, int H, int W, int Ho, int Wo) {
        int total = BATCH * C * Ho * Wo;
        maxpool2_kernel<<<(total + 255) / 256, 256, 0, stream>>>(in, out, C, H, W, Ho, Wo);
    };

    // ---------------- cnnf ----------------
    conv(x, 1, nullptr, 0,  1, nullptr, t1,  32, 36, 36, 36, 36, 1, 1, 0, 0);
    conv(t1, 32, nullptr, 0, 3, nullptr, o1,  32, 36, 36, 36, 36, 1, 1, 0, 0);
    conv(o1, 32, nullptr, 0, 5, nullptr, o2,  64, 36, 36, 36, 36, 1, 1, 0, 0);
    pool(o2, o2m, 64, 36, 36, 18, 18);
    conv(o2m, 64, nullptr, 0, 7, nullptr, o2p, 64, 18, 18, 18, 18, 1, 1, 0, 0);
    conv(o2p, 64, nullptr, 0, 9, nullptr, o3,  64, 18, 18, 18, 18, 1, 1, 0, 0);
    pool(o3, o3m, 64, 18, 18, 9, 9);
    conv(o3m, 64, nullptr, 0, 11, nullptr, o3p, 128, 9, 9, 9, 9, 1, 1, 0, 0);
    // o3u = deconv(up2(o3p), d1): logical 18x18 input (stored 9x9), no relu
    conv(o3p, 128, nullptr, 0, 13, nullptr, o3u, 64, 18, 18, 18, 18, 1, 0, 1, 1);
    // concat(o3u, o2p) -> f4a -> f4b
    conv(o3u, 64, o2p, 64, 15, nullptr, o4a, 64, 18, 18, 18, 18, 1, 1, 0, 0);
    conv(o4a, 64, nullptr, 0, 17, nullptr, o4,  64, 18, 18, 18, 18, 1, 1, 0, 0);
    // o4u = deconv(up2(o4), d2): logical 36x36 input (stored 18x18), no relu
    conv(o4, 64, nullptr, 0, 19, nullptr, o4u, 32, 36, 36, 36, 36, 1, 0, 1, 1);
    // concat(o4u, o1) -> f5a -> f5b -> f5c (E)
    conv(o4u, 32, o1, 32, 21, nullptr, f5t1, 32, 36, 36, 36, 36, 1, 1, 0, 0);
    conv(f5t1, 32, nullptr, 0, 23, nullptr, f5t2, 32, 36, 36, 36, 36, 1, 1, 0, 0);
    conv(f5t2, 32, nullptr, 0, 25, nullptr, Ebuf, 3, 36, 36, 36, 36, 1, 0, 0, 0);

    // ---------------- cnny ----------------
    conv(x, 1, nullptr, 0, 27, nullptr, yt1, 32, 36, 36, 36, 36, 1, 1, 0, 0);
    conv(yt1, 32, nullptr, 0, 29, nullptr, yt2, 32, 36, 36, 36, 36, 1, 1, 0, 0);
    conv(yt2, 32, nullptr, 0, 31, nullptr, yt1, 32, 36, 36, 36, 36, 1, 1, 0, 0);
    conv(yt1, 32, nullptr, 0, 33, x, Yv, 1, 36, 36, 36, 36, 1, 0, 0, 0); // +x residual

    // ---------------- cnnu ----------------
    conv(x, 1, nullptr, 0, 35, nullptr, cu1, 32, 36, 36, 18, 18, 2, 1, 0, 0);
    conv(cu1, 32, nullptr, 0, 37, nullptr, cu2, 32, 18, 18, 18, 18, 1, 1, 0, 0);
    pool(cu2, cu2m, 32, 18, 18, 9, 9);
    conv(cu2m, 32, nullptr, 0, 39, nullptr, cu3, 32, 9, 9, 9, 9, 1, 1, 0, 0);
    pool(cu3, cu3m, 32, 9, 9, 5, 5);                    // ceil_mode
    conv(cu3m, 32, nullptr, 0, 41, nullptr, cu4, 32, 5, 5, 5, 5, 1, 1, 0, 0);
    pool(cu4, cu4m, 32, 5, 5, 3, 3);                    // ceil_mode
    fc_kernel<<<1, 256, 0, stream>>>(cu4m, IN[43], IN[44], fco, 288, 32, 0);
    fc_kernel<<<1, 256, 0, stream>>>(fco,  IN[45], IN[46], uu,  32,  1,  1);

    // ---------------- graph + augmented system ----------------
    build_A_kernel<<<dim3(NPIX, BATCH), 256, 0, stream>>>(Ebuf, Yv, uu, Amat);

    // ---------------- blocked LU (no pivoting; A strictly diag-dominant) ----
    for (int k = 0; k < 81; ++k) {
        invert16_kernel<<<BATCH, 256, 0, stream>>>(Amat, Dinv, k);
        if (k < 80) {
            panel_kernel<<<dim3(80 - k, BATCH), 32, 0, stream>>>(Amat, Dinv, k);
            int nct = 81 - k;   // column tiles k+1 .. 81 (incl. augmented y tile)
            trailing_kernel<<<dim3((nct + 3) / 4, 80 - k, BATCH), 32, 0, stream>>>(Amat, k);
        }
    }
    backsub_kernel<<<BATCH, 256, 0, stream>>>(Amat, Dinv, (float*)d_out);
}